// SinusoidsGenerator_30837865185394
// MI455X (gfx1250) — compile-verified
//
#include <hip/hip_runtime.h>
#include <hip/hip_bf16.h>

typedef __attribute__((ext_vector_type(2))) float v2f;
typedef __attribute__((ext_vector_type(8))) float v8f;

#define T_FRAMES 500
#define HOP 240
#define L_SAMP (T_FRAMES * HOP)   // 120000
#define NYQ 12000.0f
#define N_HARM 200

// ---------------------------------------------------------------------------
// Kernel 1: per-frame f0 sums (closed form from linear upsampling) and
// exclusive prefix over frames -> P[b][t] = cumsum of f0 at frame start.
// Frame sum (with index clamping) = 30*c[t-1] + 180*c[t] + 30*c[t+1].
// ---------------------------------------------------------------------------
__global__ __launch_bounds__(256) void frame_prefix_kernel(
    const float* __restrict__ cf0, float* __restrict__ P) {
  __shared__ float S[T_FRAMES];
  const int b = blockIdx.x;
  const float* c = cf0 + b * T_FRAMES;
  for (int t = threadIdx.x; t < T_FRAMES; t += blockDim.x) {
    float c0  = c[t];
    float cm1 = (t > 0)            ? c[t - 1] : c0;
    float cp1 = (t < T_FRAMES - 1) ? c[t + 1] : c0;
    S[t] = 30.0f * cm1 + 180.0f * c0 + 30.0f * cp1;
  }
  __syncthreads();
  if (threadIdx.x == 0) {            // 500 serial adds: negligible
    float run = 0.0f;
    float* p = P + b * T_FRAMES;
    for (int t = 0; t < T_FRAMES; ++t) { p[t] = run; run += S[t]; }
  }
}

// ---------------------------------------------------------------------------
// Kernel 2: one wave = 16 consecutive samples (never crossing a frame edge).
// Within-frame inclusive cumsum in closed form; harmonics accumulated with
// V_WMMA_F32_16X16X4_F32: A = ones(16x4), B[k][n] = masked cos for sample n,
// harmonic chunk k. D rows are identical; D vgpr0 lanes 0..15 hold the 16
// per-sample sums -> one coalesced store. Harmonic indices are tracked as
// float induction variables (+4.0f/chunk, exact in fp32) so the inner loop
// has no int->float converts.
// ---------------------------------------------------------------------------
__global__ __launch_bounds__(256) void sinusoids_kernel(
    const float* __restrict__ cf0, const float* __restrict__ P,
    float* __restrict__ out) {
  const int tid   = threadIdx.x;
  const int wave  = tid >> 5;
  const int lane  = tid & 31;
  const int chunk = blockIdx.x * 8 + wave;   // 8 waves / block
  const int s0    = chunk * 16;              // first flat sample of this wave

  const int b  = s0 / L_SAMP;
  const int r  = s0 - b * L_SAMP;            // sample within batch
  const int t  = r / HOP;                    // frame
  const int j0 = r - t * HOP;                // sample within frame

  const int n  = lane & 15;                  // sample slot 0..15
  const int hk = lane >> 4;                  // half-wave: harmonic sub-chunk
  const int j  = j0 + n;

  const float* c = cf0 + b * T_FRAMES;
  float c0  = c[t];
  float cm1 = (t > 0)            ? c[t - 1] : c0;
  float cp1 = (t < T_FRAMES - 1) ? c[t + 1] : c0;
  float d1 = c0 - cm1;
  float d2 = cp1 - c0;

  // Inclusive within-frame cumsum cs(j) and instantaneous f0(j), closed form.
  // First half (j<120):  f0 = cm1 + d1*w,  w = 0.5 + (j+0.5)/240
  // Second half (j>=120): f0 = c0 + d2*v,  v = (j+0.5)/240 - 0.5
  float jf = (float)j;
  float cs, f0;
  if (j < 120) {
    float jp1 = jf + 1.0f;
    cs = jp1 * cm1 + d1 * (0.5f * jp1 + jp1 * jp1 * (1.0f / 480.0f));
    f0 = cm1 + d1 * (0.5f + (jf + 0.5f) * (1.0f / 240.0f));
  } else {
    float m = jf - 119.0f;
    cs = 120.0f * cm1 + 90.0f * d1 + m * c0 + d2 * (m * m * (1.0f / 480.0f));
    f0 = c0 + d2 * ((jf + 0.5f) * (1.0f / 240.0f) - 0.5f);
  }

  // Phase in revolutions per harmonic index: cos(2*pi * u * k)
  float u = (P[b * T_FRAMES + t] + cs) * (1.0f / 24000.0f);

  v8f acc = {0.f, 0.f, 0.f, 0.f, 0.f, 0.f, 0.f, 0.f};
  v2f ones = {1.0f, 1.0f};

  // Float induction: fkA = first harmonic index handled by this lane,
  // kfirst = smallest harmonic index in the current 4-chunk (for the break).
  float fkA    = 1.0f + 2.0f * (float)hk;    // lanes 0-15: k,k+1 ; 16-31: k+2,k+3
  float kfirst = 1.0f;
  for (int it = 0; it < N_HARM / 4; ++it) {
    // Mask is monotone in k; wave-uniform early exit once the whole chunk
    // (and all later ones) is above Nyquist for every sample in this wave.
    if (!__any(f0 * kfirst < NYQ)) break;

    float fk1 = fkA;
    float fk2 = fkA + 1.0f;
    float b0 = (f0 * fk1 < NYQ)
                   ? __builtin_amdgcn_cosf(__builtin_amdgcn_fractf(u * fk1))
                   : 0.0f;
    float b1 = (f0 * fk2 < NYQ)
                   ? __builtin_amdgcn_cosf(__builtin_amdgcn_fractf(u * fk2))
                   : 0.0f;
    v2f bv = {b0, b1};
    // D = ones(16x4) * B(4x16) + D  -> every row of D = per-sample harmonic sums
    acc = __builtin_amdgcn_wmma_f32_16x16x4_f32(
        /*neg_a=*/false, ones, /*neg_b=*/false, bv,
        /*c_mod=*/(short)0, acc, /*reuse_a=*/false, /*reuse_b=*/false);

    fkA    += 4.0f;
    kfirst += 4.0f;
  }

  // D vgpr0: lanes 0-15 = D[0][n] = sum for sample n (lanes 16-31 duplicate row 8)
  if (lane < 16) out[s0 + n] = 0.1f * acc[0];
}

extern "C" void kernel_launch(void* const* d_in, const int* in_sizes, int n_in,
                              void* d_out, int out_size, void* d_ws, size_t ws_size,
                              hipStream_t stream) {
  const float* cf0 = (const float*)d_in[0];
  float* out = (float*)d_out;
  float* P   = (float*)d_ws;                 // B * 500 floats of scratch

  int B = in_sizes[0] / T_FRAMES;            // (B, 500, 1) -> B

  frame_prefix_kernel<<<B, 256, 0, stream>>>(cf0, P);

  int waves  = (B * L_SAMP) / 16;            // 16 samples per wave
  int blocks = waves / 8;                    // 8 waves per 256-thread block
  sinusoids_kernel<<<blocks, 256, 0, stream>>>(cf0, P, out);
}